// JointAttention_1752346657094
// MI455X (gfx1250) — compile-verified
//
#include <hip/hip_runtime.h>
#include <hip/hip_bf16.h>
#include <math.h>

// ---------------- problem constants (from reference) ----------------
#define MODEL 1024
#define NH    16
#define HD    64
#define BB    2
#define SS    2048
#define LLAT  256
#define LT    512
#define LSP   512
#define LTOT  (SS + LLAT + LT + LSP)   // 3328
#define NEG_BIG (-1e30f)

typedef __attribute__((ext_vector_type(2))) float v2f;
typedef __attribute__((ext_vector_type(4))) float v4f;
typedef __attribute__((ext_vector_type(8))) float v8f;
typedef __attribute__((ext_vector_type(4))) int   v4i;

// D(16x16,f32) = A(16x4,f32) x B(4x16,f32) + C  -- exact fp32 on matrix pipes
__device__ __forceinline__ v8f wmma_f32(v2f a, v2f b, v8f c) {
  return __builtin_amdgcn_wmma_f32_16x16x4_f32(
      /*neg_a=*/false, a, /*neg_b=*/false, b,
      /*c_mod=*/(short)0, c, /*reuse_a=*/false, /*reuse_b=*/false);
}

// ---------------- CDNA5 async global->LDS copy (ASYNCcnt path) ----------------
#if defined(__has_builtin)
#if __has_builtin(__builtin_amdgcn_global_load_async_to_lds_b128)
#define HAS_ASYNC_LDS 1
#endif
#endif

#ifdef HAS_ASYNC_LDS
__device__ __forceinline__ void async_cp16(const float* g, float* l) {
  // builtin signature (per hipcc diagnostic): generic int4* src, int4* lds-dst,
  // imm offset, imm cpol
  __builtin_amdgcn_global_load_async_to_lds_b128((v4i*)g, (v4i*)l,
                                                 /*offset=*/0, /*cpol=*/0);
}
#if __has_builtin(__builtin_amdgcn_s_wait_asynccnt)
#define WAIT_ASYNCCNT(n) __builtin_amdgcn_s_wait_asynccnt(n)
#else
#define WAIT_ASYNCCNT(n) asm volatile("s_wait_asynccnt %0" ::"n"(n) : "memory")
#endif
#else
__device__ __forceinline__ void async_cp16(const float* g, float* l) {
  *(v4f*)l = *(const v4f*)g;   // synchronous fallback
}
#define WAIT_ASYNCCNT(n) ((void)0)
#endif

// =====================================================================
// fp32 WMMA GEMM:  C[4096,1024] = A[4096,1024] @ W[1024,1024]
// Block = 256 threads = 8 waves; block tile 128x64; wave tile 32x32.
// K staged through double-buffered LDS in steps of 16 via async copies.
// =====================================================================
__global__ __launch_bounds__(256)
void gemm_f32_wmma(const float* __restrict__ A, const float* __restrict__ W,
                   float* __restrict__ C) {
  __shared__ __align__(16) float As[2][128][20];  // rows padded: 80B, 16B-mult,
                                                  // stride 20 banks -> conflict free
  __shared__ __align__(16) float Bs[2][16][68];   // 272B rows (16B multiple)

  const int tid  = threadIdx.x;
  const int wave = tid >> 5, lane = tid & 31;
  const int hf = lane >> 4, l = lane & 15;
  const int mw = (wave >> 1) * 32;      // wave row offset in 128-row tile
  const int nw = (wave & 1) * 32;       // wave col offset in 64-col tile
  const int m0 = blockIdx.x * 128;
  const int n0 = blockIdx.y * 64;

  // cooperative-copy coordinates (16B chunks)
  const int ar = tid >> 2,  ac = (tid & 3) * 4;    // A: 2 chunks/thread
  const int br = tid >> 4,  bc = (tid & 15) * 4;   // B: 1 chunk/thread

  v8f a00 = {}, a01 = {}, a10 = {}, a11 = {};

  auto issue = [&](int st) {
    const int k0 = st * 16, bf = st & 1;
    async_cp16(&A[(size_t)(m0 + ar) * MODEL + k0 + ac],      &As[bf][ar][ac]);
    async_cp16(&A[(size_t)(m0 + 64 + ar) * MODEL + k0 + ac], &As[bf][64 + ar][ac]);
    async_cp16(&W[(size_t)(k0 + br) * MODEL + n0 + bc],      &Bs[bf][br][bc]);
  };

  issue(0);
  const int NSTEP = MODEL / 16;   // 64
  for (int st = 0; st < NSTEP; ++st) {
    if (st + 1 < NSTEP) { issue(st + 1); WAIT_ASYNCCNT(3); }
    else                {                WAIT_ASYNCCNT(0); }
    __syncthreads();

    const int bf = st & 1;
    #pragma unroll
    for (int kk = 0; kk < 4; ++kk) {
      const int ks = 4 * kk + 2 * hf;
      v2f fa0 = *(const v2f*)&As[bf][mw + l][ks];        // ds_load_b64
      v2f fa1 = *(const v2f*)&As[bf][mw + 16 + l][ks];
      v2f fb0; fb0.x = Bs[bf][ks][nw + l];      fb0.y = Bs[bf][ks + 1][nw + l];
      v2f fb1; fb1.x = Bs[bf][ks][nw + 16 + l]; fb1.y = Bs[bf][ks + 1][nw + 16 + l];
      a00 = wmma_f32(fa0, fb0, a00);
      a01 = wmma_f32(fa0, fb1, a01);
      a10 = wmma_f32(fa1, fb0, a10);
      a11 = wmma_f32(fa1, fb1, a11);
    }
    __syncthreads();
  }

  #pragma unroll
  for (int r = 0; r < 8; ++r) {
    const size_t r0 = (size_t)(m0 + mw + r + 8 * hf);
    const size_t r1 = r0 + 16;
    C[r0 * MODEL + n0 + nw + l]      = a00[r];
    C[r0 * MODEL + n0 + nw + 16 + l] = a01[r];
    C[r1 * MODEL + n0 + nw + l]      = a10[r];
    C[r1 * MODEL + n0 + nw + 16 + l] = a11[r];
  }
}

// =====================================================================
// Per-(b,s,h) post-processing: RMS-norm on q/k, RoPE on first NH/2
// heads, scatter self k/v into concatenated caches.  One wave per row.
// =====================================================================
__global__ __launch_bounds__(256)
void qkv_post(const float* __restrict__ qraw, const float* __restrict__ kraw,
              const float* __restrict__ vraw,
              const float* __restrict__ qnw, const float* __restrict__ knw,
              const float* __restrict__ fcos, const float* __restrict__ fsin,
              const int* __restrict__ start_pos_p,
              float* __restrict__ Q, float* __restrict__ Kc, float* __restrict__ Vc) {
  const int gid  = blockIdx.x * 8 + (threadIdx.x >> 5);
  const int lane = threadIdx.x & 31;
  const int h = gid % NH;
  const int bs = gid / NH;
  const int s = bs % SS, b = bs / SS;

  const size_t base = (size_t)(b * SS + s) * MODEL + h * HD;
  float q0 = qraw[base + lane], q1 = qraw[base + lane + 32];
  float k0 = kraw[base + lane], k1 = kraw[base + lane + 32];
  const float v0 = vraw[base + lane], v1 = vraw[base + lane + 32];

  float ssq = q0 * q0 + q1 * q1;
  float ssk = k0 * k0 + k1 * k1;
  #pragma unroll
  for (int m = 16; m >= 1; m >>= 1) {
    ssq += __shfl_xor(ssq, m, 32);
    ssk += __shfl_xor(ssk, m, 32);
  }
  const float rq = rsqrtf(ssq * (1.0f / HD) + 1e-5f);
  const float rk = rsqrtf(ssk * (1.0f / HD) + 1e-5f);
  q0 *= rq * qnw[h * HD + lane]; q1 *= rq * qnw[h * HD + lane + 32];
  k0 *= rk * knw[h * HD + lane]; k1 *= rk * knw[h * HD + lane + 32];

  if (h < NH / 2) {
    const int pos = start_pos_p[0] + s;
    const float c0 = fcos[pos * 32 + (lane >> 1)];
    const float s0 = fsin[pos * 32 + (lane >> 1)];
    const float c1 = fcos[pos * 32 + ((lane + 32) >> 1)];
    const float s1 = fsin[pos * 32 + ((lane + 32) >> 1)];
    const float qp0 = __shfl_xor(q0, 1, 32), qp1 = __shfl_xor(q1, 1, 32);
    const float kp0 = __shfl_xor(k0, 1, 32), kp1 = __shfl_xor(k1, 1, 32);
    if ((lane & 1) == 0) {   // real part: r*c - i*s
      q0 = q0 * c0 - qp0 * s0;  q1 = q1 * c1 - qp1 * s1;
      k0 = k0 * c0 - kp0 * s0;  k1 = k1 * c1 - kp1 * s1;
    } else {                 // imag part: r*s + i*c
      q0 = qp0 * s0 + q0 * c0;  q1 = qp1 * s1 + q1 * c1;
      k0 = kp0 * s0 + k0 * c0;  k1 = kp1 * s1 + k1 * c1;
    }
  }

  const size_t qoff = ((size_t)(b * NH + h) * SS + s) * HD;
  Q[qoff + lane] = q0; Q[qoff + lane + 32] = q1;
  const size_t koff = ((size_t)(b * NH + h) * LTOT + s) * HD;
  Kc[koff + lane] = k0; Kc[koff + lane + 32] = k1;
  Vc[koff + lane] = v0; Vc[koff + lane + 32] = v1;
}

// =====================================================================
// Copy latent/text/speaker KV caches into concatenated [B][H][LTOT][HD]
// =====================================================================
__global__ __launch_bounds__(256)
void cache_assemble(const float* __restrict__ lat_k, const float* __restrict__ lat_v,
                    const float* __restrict__ txt_k, const float* __restrict__ txt_v,
                    const float* __restrict__ spk_k, const float* __restrict__ spk_v,
                    float* __restrict__ Kc, float* __restrict__ Vc) {
  const int gid  = blockIdx.x * 8 + (threadIdx.x >> 5);
  const int lane = threadIdx.x & 31;
  const int h = gid % NH;
  const int t = gid / NH;
  const int j = t % (LLAT + LT + LSP);
  const int b = t / (LLAT + LT + LSP);

  const float *ks, *vs; int jj, L, pos;
  if (j < LLAT)            { ks = lat_k; vs = lat_v; jj = j;             L = LLAT; pos = SS + j; }
  else if (j < LLAT + LT)  { ks = txt_k; vs = txt_v; jj = j - LLAT;      L = LT;   pos = SS + LLAT + jj; }
  else                     { ks = spk_k; vs = spk_v; jj = j - LLAT - LT; L = LSP;  pos = SS + LLAT + LT + jj; }

  const size_t src = ((size_t)(b * L + jj) * NH + h) * HD;
  const size_t dst = ((size_t)(b * NH + h) * LTOT + pos) * HD;
  Kc[dst + lane] = ks[src + lane]; Kc[dst + lane + 32] = ks[src + lane + 32];
  Vc[dst + lane] = vs[src + lane]; Vc[dst + lane + 32] = vs[src + lane + 32];
}

// ---------------- additive mask bias per (b, key position) ----------------
__global__ void build_bias(const unsigned char* __restrict__ text_mask,
                           const unsigned char* __restrict__ spk_mask,
                           const int* __restrict__ start_pos_p,
                           float* __restrict__ bias) {
  const int i = blockIdx.x * blockDim.x + threadIdx.x;
  if (i >= BB * LTOT) return;
  const int b = i / LTOT, p = i % LTOT;
  float v;
  if (p < SS)                       v = 0.f;
  else if (p < SS + LLAT)           v = ((p - SS) * 4 < start_pos_p[0]) ? 0.f : NEG_BIG;
  else if (p < SS + LLAT + LT)      v = text_mask[b * LT + (p - SS - LLAT)] ? 0.f : NEG_BIG;
  else                              v = spk_mask[b * LSP + (p - SS - LLAT - LT)] ? 0.f : NEG_BIG;
  bias[i] = v;
}

// =====================================================================
// Flash-style attention.  Block = 8 waves sharing one (b,h); each wave
// owns a 16-row Q tile.  K/V 16x64 tiles are staged cooperatively into
// double-buffered LDS via CDNA5 async copies (shared by all 8 waves),
// fp32 WMMA for Q·K^T and P·V, fused 1/l + sigmoid(gate) epilogue.
// =====================================================================
__global__ __launch_bounds__(256)
void attention_fwd(const float* __restrict__ Q, const float* __restrict__ Kc,
                   const float* __restrict__ Vc, const float* __restrict__ bias,
                   const float* __restrict__ gate, float* __restrict__ Og) {
  __shared__ __align__(16) float Ks[2][16][68];
  __shared__ __align__(16) float Vs[2][16][68];
  __shared__ float Pl[8][16][17];   // per-wave P staging (C-layout -> A-layout)

  const int tid  = threadIdx.x;
  const int wave = tid >> 5, lane = tid & 31;
  const int hf = lane >> 4, l = lane & 15;
  const int b = blockIdx.z, h = blockIdx.y;
  const int qt = blockIdx.x * 8 + wave;          // 16-row q tile index

  const float* Qb = Q  + ((size_t)(b * NH + h) * SS + qt * 16) * HD;
  const float* Kb = Kc + (size_t)(b * NH + h) * LTOT * HD;
  const float* Vb = Vc + (size_t)(b * NH + h) * LTOT * HD;
  const float* biasb = bias + b * LTOT;

  // preload Q fragments: 16 K-slices of 4 over HD=64
  v2f qf[16];
  #pragma unroll
  for (int ks = 0; ks < 16; ++ks) {
    qf[ks].x = Qb[l * HD + 4 * ks + 2 * hf];
    qf[ks].y = Qb[l * HD + 4 * ks + 2 * hf + 1];
  }

  v8f acc[4] = {{}, {}, {}, {}};
  float mrow[8], lrow[8];
  #pragma unroll
  for (int r = 0; r < 8; ++r) { mrow[r] = -INFINITY; lrow[r] = 0.f; }
  const float scale = 0.125f;   // 1/sqrt(64)

  const int kr = tid >> 4, kc = (tid & 15) * 4;  // staging coords (16B chunks)
  auto issueKV = [&](int kb) {
    const int bf = kb & 1, kpos = kb * 16;
    async_cp16(&Kb[(size_t)(kpos + kr) * HD + kc], &Ks[bf][kr][kc]);
    async_cp16(&Vb[(size_t)(kpos + kr) * HD + kc], &Vs[bf][kr][kc]);
  };

  issueKV(0);
  const int NB = LTOT / 16;   // 208
  for (int kb = 0; kb < NB; ++kb) {
    const int kpos = kb * 16, bf = kb & 1;
    if (kb + 1 < NB) { issueKV(kb + 1); WAIT_ASYNCCNT(2); }
    else             {                  WAIT_ASYNCCNT(0); }
    __syncthreads();

    // scores S(16x16) = Q(16x64) @ K^T(64x16)
    v8f s = {};
    #pragma unroll
    for (int ks = 0; ks < 16; ++ks) {
      v2f kf = *(const v2f*)&Ks[bf][l][4 * ks + 2 * hf];   // ds_load_b64
      s = wmma_f32(qf[ks], kf, s);
    }

    const float bb = biasb[kpos + l];   // col-wise additive mask

    // online softmax, rows striped across 16-lane halves
    #pragma unroll
    for (int r = 0; r < 8; ++r) {
      float v = s[r] * scale + bb;
      float rmax = v;
      #pragma unroll
      for (int m = 1; m < 16; m <<= 1) rmax = fmaxf(rmax, __shfl_xor(rmax, m, 16));
      const float mn = fmaxf(mrow[r], rmax);
      const float p = __expf(v - mn);
      float psum = p;
      #pragma unroll
      for (int m = 1; m < 16; m <<= 1) psum += __shfl_xor(psum, m, 16);
      const float alpha = __expf(mrow[r] - mn);
      lrow[r] = lrow[r] * alpha + psum;
      mrow[r] = mn;
      #pragma unroll
      for (int t = 0; t < 4; ++t) acc[t][r] *= alpha;
      Pl[wave][r + 8 * hf][l] = p;     // C-layout -> LDS
    }

    // O(16x64) += P(16x16) @ V(16x64)
    #pragma unroll
    for (int kk = 0; kk < 4; ++kk) {
      v2f pa;
      pa.x = Pl[wave][l][4 * kk + 2 * hf];
      pa.y = Pl[wave][l][4 * kk + 2 * hf + 1];
      #pragma unroll
      for (int t = 0; t < 4; ++t) {
        v2f vf;
        vf.x = Vs[bf][4 * kk + 2 * hf][16 * t + l];
        vf.y = Vs[bf][4 * kk + 2 * hf + 1][16 * t + l];
        acc[t] = wmma_f32(pa, vf, acc[t]);
      }
    }
    __syncthreads();
  }

  // epilogue: normalize + sigmoid gating, layout [b][s][h*64+d]
  #pragma unroll
  for (int r = 0; r < 8; ++r) {
    const float inv = 1.f / lrow[r];
    const int row = qt * 16 + r + 8 * hf;
    const size_t gbase = (size_t)(b * SS + row) * MODEL + h * HD;
    #pragma unroll
    for (int t = 0; t < 4; ++t) {
      const int d = 16 * t + l;
      const float g = gate[gbase + d];
      Og[gbase + d] = acc[t][r] * inv * (1.f / (1.f + __expf(-g)));
    }
  }
}

// =====================================================================
extern "C" void kernel_launch(void* const* d_in, const int* in_sizes, int n_in,
                              void* d_out, int out_size, void* d_ws, size_t ws_size,
                              hipStream_t stream) {
  (void)in_sizes; (void)n_in; (void)out_size; (void)ws_size;
  const float* x            = (const float*)d_in[0];
  const unsigned char* tmsk = (const unsigned char*)d_in[1];  // jnp.bool_ -> 1 byte
  const unsigned char* smsk = (const unsigned char*)d_in[2];
  const float* fcos         = (const float*)d_in[3];
  const float* fsin         = (const float*)d_in[4];
  const float* kv_text_k    = (const float*)d_in[5];
  const float* kv_text_v    = (const float*)d_in[6];
  const float* kv_spk_k     = (const float*)d_in[7];
  const float* kv_spk_v     = (const float*)d_in[8];
  const float* kv_lat_k     = (const float*)d_in[9];
  const float* kv_lat_v     = (const float*)d_in[10];
  const int*   start_pos    = (const int*)d_in[11];
  const float* wq           = (const float*)d_in[12];
  const float* wk           = (const float*)d_in[13];
  const float* wv           = (const float*)d_in[14];
  const float* gate_w       = (const float*)d_in[15];
  const float* wo           = (const float*)d_in[16];
  const float* q_norm_w     = (const float*)d_in[17];
  const float* k_norm_w     = (const float*)d_in[18];

  const int M = BB * SS;                 // 4096 rows
  float* ws = (float*)d_ws;
  size_t off = 0;
  float* q_raw  = ws + off; off += (size_t)M * MODEL;
  float* k_raw  = ws + off; off += (size_t)M * MODEL;
  float* v_raw  = ws + off; off += (size_t)M * MODEL;
  float* gate   = ws + off; off += (size_t)M * MODEL;
  float* Qbuf   = ws + off; off += (size_t)BB * NH * SS * HD;
  float* Kcat   = ws + off; off += (size_t)BB * NH * LTOT * HD;
  float* Vcat   = ws + off; off += (size_t)BB * NH * LTOT * HD;
  float* bias   = ws + off; off += (size_t)BB * LTOT;
  float* attn_g = ws + off; off += (size_t)M * MODEL;

  dim3 gg(M / 128, MODEL / 64), gb(256);
  gemm_f32_wmma<<<gg, gb, 0, stream>>>(x, wq,     q_raw);
  gemm_f32_wmma<<<gg, gb, 0, stream>>>(x, wk,     k_raw);
  gemm_f32_wmma<<<gg, gb, 0, stream>>>(x, wv,     v_raw);
  gemm_f32_wmma<<<gg, gb, 0, stream>>>(x, gate_w, gate);

  qkv_post<<<(BB * SS * NH) / 8, 256, 0, stream>>>(
      q_raw, k_raw, v_raw, q_norm_w, k_norm_w, fcos, fsin, start_pos,
      Qbuf, Kcat, Vcat);

  cache_assemble<<<(BB * (LLAT + LT + LSP) * NH) / 8, 256, 0, stream>>>(
      kv_lat_k, kv_lat_v, kv_text_k, kv_text_v, kv_spk_k, kv_spk_v, Kcat, Vcat);

  build_bias<<<(BB * LTOT + 255) / 256, 256, 0, stream>>>(tmsk, smsk, start_pos, bias);

  attention_fwd<<<dim3(SS / (16 * 8), NH, BB), 256, 0, stream>>>(
      Qbuf, Kcat, Vcat, bias, gate, attn_g);

  gemm_f32_wmma<<<gg, gb, 0, stream>>>(attn_g, wo, (float*)d_out);
}